// Environment_28183575396882
// MI455X (gfx1250) — compile-verified
//
#include <hip/hip_runtime.h>
#include <hip/hip_bf16.h>
#include <stdint.h>

// Grid geometry is fixed by the reference: H = W = 64 -> 4096 cells/env.
#define CELLS         4096
#define WORDS         128          // CELLS / 32 bits
#define GRID_W        64
#define ENVS_PER_BLK  128
#define PAIR_STRIDE_W 258          // 256 data words (128 wall/goal pairs) + 2 pad words (8B align, de-skews banks)

// ---------------------------------------------------------------------------
// Kernel A: pack world floats into bitmaps.  walls[c]==1.0 -> wall bit,
// goals[c]==10.0 -> goal bit.  wave32 ballot makes lane l supply bit l of
// word w, exactly matching the rollout lookup (word = idx>>5, bit = idx&31).
// World is 256MB (> L2's 192MB) and read exactly once -> non-temporal loads.
// ---------------------------------------------------------------------------
__global__ void pack_world_kernel(const float* __restrict__ world,
                                  unsigned* __restrict__ wallsP,
                                  unsigned* __restrict__ goalsP,
                                  int B) {
    const int b = blockIdx.x;
    if (b >= B) return;
    const int lane   = threadIdx.x & 31;
    const int wave   = threadIdx.x >> 5;
    const int nwaves = blockDim.x >> 5;
    const float* wallsSrc = world + (size_t)b * 2 * CELLS;
    const float* goalsSrc = wallsSrc + CELLS;
    for (int w = wave; w < WORDS; w += nwaves) {
        const float wv = __builtin_nontemporal_load(&wallsSrc[w * 32 + lane]);
        const float gv = __builtin_nontemporal_load(&goalsSrc[w * 32 + lane]);
        const unsigned wb = __builtin_amdgcn_ballot_w32(wv == 1.0f);
        const unsigned gb = __builtin_amdgcn_ballot_w32(gv == 10.0f);
        if (lane == 0) {
            wallsP[(size_t)b * WORDS + w] = wb;
            goalsP[(size_t)b * WORDS + w] = gb;
        }
    }
}

// ---------------------------------------------------------------------------
// Kernel B: one thread per environment.  Bitmaps for the block's 128 envs are
// pulled into LDS with gfx1250 async global->LDS DMA (ASYNCcnt tracked),
// interleaved as (wall word, goal word) pairs so each rollout step needs a
// single aligned ds_load_b64.  State is kept in integers to minimize the
// per-step dependent-chain latency (only 256 waves exist machine-wide, so the
// whole kernel is one long latency chain).
// ---------------------------------------------------------------------------
__global__ void rollout_kernel(const float* __restrict__ s0,
                               const int* __restrict__ a,
                               const unsigned* __restrict__ wallsP,
                               const unsigned* __restrict__ goalsP,
                               float* __restrict__ sOut,
                               float* __restrict__ rOut,
                               int B, int T) {
    extern __shared__ unsigned smem[];
    const int t  = threadIdx.x;                      // 0..127
    const int e0 = blockIdx.x * ENVS_PER_BLK;
    int envs = B - e0;
    if (envs > ENVS_PER_BLK) envs = ENVS_PER_BLK;

    // Low 32 bits of a flat LDS pointer are the wave-relative LDS byte address
    // (ISA 10.2) — same convention async-to-LDS VDST uses.
    const unsigned ldsBase = (unsigned)(uintptr_t)smem;

    // --- async copy: thread t fetches word t of every env's wall+goal maps ---
    {
        const uint64_t wBase = (uint64_t)(wallsP + (size_t)e0 * WORDS);
        const uint64_t gBase = (uint64_t)(goalsP + (size_t)e0 * WORDS);
        for (int i = 0; i < envs; ++i) {
            const unsigned goff = (unsigned)((i * WORDS + t) * 4);
            const unsigned lW   = ldsBase + (unsigned)(i * PAIR_STRIDE_W * 4 + t * 8);
            const unsigned lG   = lW + 4;
            asm volatile("global_load_async_to_lds_b32 %0, %1, %2"
                         :: "v"(lW), "v"(goff), "s"(wBase) : "memory");
            asm volatile("global_load_async_to_lds_b32 %0, %1, %2"
                         :: "v"(lG), "v"(goff), "s"(gBase) : "memory");
        }
        asm volatile("s_wait_asynccnt 0" ::: "memory");
    }
    __syncthreads();
    if (t >= envs) return;

    const int b = e0 + t;
    const uint64_t* myMap =
        (const uint64_t*)((const char*)smem + (size_t)t * PAIR_STRIDE_W * 4);

    // Positions are always exact small integers: track as int (shorter VALU chain).
    int ii = (int)s0[b * 2 + 0];
    int jj = (int)s0[b * 2 + 1];
    const int T1 = T + 1;
    float* sRow = sOut + (size_t)b * T1 * 2;
    float* rRow = rOut + (size_t)b * T1;
    sRow[0] = (float)ii; sRow[1] = (float)jj; rRow[0] = 0.0f;

    const int* ab = a + (size_t)b * T;

    auto step = [&](int at, int k) {
        const int di = (at == 2) - (at == 1);                  // 1:-row, 2:+row
        const int dj = (at == 4) - (at == 3);                  // 3:-col, 4:+col
        int ni = ii + di, nj = jj + dj;
        int idx = ni * GRID_W + nj;
        idx = idx < 0 ? 0 : (idx > (CELLS - 1) ? (CELLS - 1) : idx);  // JAX clip-mode gather
        const uint64_t pair = myMap[idx >> 5];                 // one ds_load_b64: wall|goal words
        const unsigned bit  = (unsigned)idx & 31u;
        const bool hitGoal  = ((unsigned)(pair >> 32) >> bit) & 1u;
        const bool hitWall  = (!hitGoal) && ((((unsigned)pair) >> bit) & 1u);
        const float r = hitGoal ? 1.0f : (hitWall ? -1.0f : -0.01f);
        if (hitWall) { ni = ii; nj = jj; }                     // revert move on wall hit
        ii = ni; jj = nj;
        // Trajectory is write-once, never re-read: NT store, packed as one b64.
        union { float f[2]; uint64_t u; } sv;
        sv.f[0] = (float)ii; sv.f[1] = (float)jj;
        __builtin_nontemporal_store(sv.u, (uint64_t*)(sRow + (size_t)(k + 1) * 2));
        __builtin_nontemporal_store(r, &rRow[k + 1]);
    };

    for (int st = 0; st < T; st += 4) {
        if ((st & 31) == 0) __builtin_prefetch(ab + st + 64, 0, 0);  // global_prefetch_b8
        const int4 av = *(const int4*)(ab + st);               // 16B-aligned action chunk
        step(av.x, st + 0);
        step(av.y, st + 1);
        step(av.z, st + 2);
        step(av.w, st + 3);
    }
}

// ---------------------------------------------------------------------------
// Launch: inputs are s0 [B,2] f32, a [B,T] i32, world [B,2,64,64] f32, T.
// d_out = s_full [B,T+1,2] f32  followed by  r_full [B,T+1] f32.
// d_ws holds the packed bitmaps (B*128 u32 walls + B*128 u32 goals = 8 MB).
// ---------------------------------------------------------------------------
extern "C" void kernel_launch(void* const* d_in, const int* in_sizes, int n_in,
                              void* d_out, int out_size, void* d_ws, size_t ws_size,
                              hipStream_t stream) {
    const float* s0    = (const float*)d_in[0];
    const int*   a     = (const int*)d_in[1];
    const float* world = (const float*)d_in[2];

    const int B = in_sizes[0] / 2;
    const int T = in_sizes[1] / B;

    unsigned* wallsP = (unsigned*)d_ws;
    unsigned* goalsP = wallsP + (size_t)B * WORDS;

    float* sOut = (float*)d_out;
    float* rOut = sOut + (size_t)B * (T + 1) * 2;

    // Phase 1: bandwidth phase — 8192 blocks stream the 256MB world once (NT).
    pack_world_kernel<<<dim3(B), dim3(256), 0, stream>>>(world, wallsP, goalsP, B);

    // Phase 2: latency phase — 1 thread/env, bitmaps resident in LDS.
    const int blocks = (B + ENVS_PER_BLK - 1) / ENVS_PER_BLK;
    const size_t shmem = (size_t)ENVS_PER_BLK * PAIR_STRIDE_W * sizeof(unsigned); // ~129 KB
    hipFuncSetAttribute((const void*)rollout_kernel,
                        hipFuncAttributeMaxDynamicSharedMemorySize, (int)shmem);
    rollout_kernel<<<dim3(blocks), dim3(ENVS_PER_BLK), shmem, stream>>>(
        s0, a, wallsP, goalsP, sOut, rOut, B, T);
}